// FeatureTimeAttentionLayer_8890582302972
// MI455X (gfx1250) — compile-verified
//
#include <hip/hip_runtime.h>

// Problem sizes (fixed by the reference)
#define BS_    256
#define T_     64
#define DF_    32
#define UNITS_ 128

#define FP_STRIDE 132   // padded LDS row stride for fp (floats): 16B-aligned rows, conflict-free WMMA tile reads
#define G_STRIDE  36    // augmented 32x36 system (col 32 = rhs); 16B-aligned rows for float4 register loads

typedef __attribute__((ext_vector_type(2))) float v2f;   // f32 WMMA A/B operand: 16x4 = 64 f32 / 32 lanes = 2 VGPRs
typedef __attribute__((ext_vector_type(8))) float v8f;   // f32 WMMA C/D: 16x16 f32 = 8 VGPRs

__global__ __launch_bounds__(256) void fta_zero_kernel(float* __restrict__ out, int n) {
    int i = blockIdx.x * blockDim.x + threadIdx.x;
    if (i < n) out[i] = 0.0f;
}

// One wave (32 lanes) per (b,t). Grid = BS*T = 16384 blocks.
__global__ __launch_bounds__(32) void fta_kernel(
    const float* __restrict__ inputs,   // (BS,T,DF)
    const float* __restrict__ latent,   // (BS,T,UNITS)
    const float* __restrict__ wkernel,  // (DF,UNITS)
    const float* __restrict__ wbias,    // (DF,UNITS)
    const float* __restrict__ beta,     // (T)
    float* __restrict__ z)              // (BS,UNITS) accumulated via f32 atomics
{
    __shared__ float fp_lds[DF_ * FP_STRIDE];  // fp matrix, row-major, padded
    __shared__ float G[DF_ * G_STRIDE];        // augmented [A | rhs]
    __shared__ float pivrow[G_STRIDE];         // broadcast buffer for the pivot row
    __shared__ float y_lds[UNITS_];
    __shared__ float x_lds[DF_];
    __shared__ float alpha[DF_];

    const int lane = threadIdx.x;          // 0..31
    const int bt   = blockIdx.x;           // 0..BS*T-1
    const int b    = bt / T_;
    const int t    = bt - b * T_;

    // ---- stage x (32) and y (128) into LDS (coalesced global loads) ----
    x_lds[lane] = inputs[(size_t)bt * DF_ + lane];
    reinterpret_cast<float4*>(y_lds)[lane] =
        reinterpret_cast<const float4*>(latent + (size_t)bt * UNITS_)[lane];
    __syncthreads();

    // ---- fp[d,:] = x[d]*kernel[d,:] + bias[d,:]  (lane owns 4 units per row) ----
    #pragma unroll 4
    for (int d = 0; d < DF_; ++d) {
        const float xd = x_lds[d];
        float4 k4 = reinterpret_cast<const float4*>(wkernel + d * UNITS_)[lane];
        float4 b4 = reinterpret_cast<const float4*>(wbias   + d * UNITS_)[lane];
        float4 f4;
        f4.x = fmaf(xd, k4.x, b4.x);
        f4.y = fmaf(xd, k4.y, b4.y);
        f4.z = fmaf(xd, k4.z, b4.z);
        f4.w = fmaf(xd, k4.w, b4.w);
        *reinterpret_cast<float4*>(fp_lds + d * FP_STRIDE + 4 * lane) = f4;
    }
    __syncthreads();

    // ---- rhs[d=lane] = fp[d,:] . y  -> augmented column 32 ----
    {
        const float4* frow = reinterpret_cast<const float4*>(fp_lds + lane * FP_STRIDE);
        const float4* yv   = reinterpret_cast<const float4*>(y_lds);
        float acc = 0.0f;
        #pragma unroll
        for (int u = 0; u < UNITS_ / 4; ++u) {
            float4 f = frow[u];
            float4 yy = yv[u];
            acc = fmaf(f.x, yy.x, acc);
            acc = fmaf(f.y, yy.y, acc);
            acc = fmaf(f.z, yy.z, acc);
            acc = fmaf(f.w, yy.w, acc);
        }
        G[lane * G_STRIDE + DF_] = acc;
    }

    // ---- Gram A = fp * fp^T via V_WMMA_F32_16X16X4_F32, exploiting symmetry ----
    // A-tile layout: lane = M (mod 16), K = vgpr + 2*(lane>=16). The identical register
    // layout serves as the B operand of fp^T. G10 = G01^T is NOT computed (mirrored on scatter).
    v8f g00 = {}, g01 = {}, g11 = {};
    const int half = lane >> 4;     // 0: K, K+1   1: K+2, K+3
    const int r    = lane & 15;     // tile row / col index
    const float* base0 = fp_lds + r        * FP_STRIDE + 2 * half;  // fp rows 0..15
    const float* base1 = fp_lds + (16 + r) * FP_STRIDE + 2 * half;  // fp rows 16..31
    #pragma unroll 8
    for (int k0 = 0; k0 < UNITS_; k0 += 4) {
        v2f a0 = *reinterpret_cast<const v2f*>(base0 + k0);   // ds_load_b64
        v2f a1 = *reinterpret_cast<const v2f*>(base1 + k0);   // ds_load_b64
        // (neg_a, A, neg_b, B, c_mod, C, reuse_a, reuse_b)
        g00 = __builtin_amdgcn_wmma_f32_16x16x4_f32(false, a0, false, a0, (short)0, g00, false, false);
        g01 = __builtin_amdgcn_wmma_f32_16x16x4_f32(false, a0, false, a1, (short)0, g01, false, false);
        g11 = __builtin_amdgcn_wmma_f32_16x16x4_f32(false, a1, false, a1, (short)0, g11, false, false);
    }

    // ---- scatter C/D tiles into the augmented LDS system (+ mirrored g01 for G10) ----
    // C/D layout: vgpr v -> M = v + 8*(lane>=16), N = lane%16
    #pragma unroll
    for (int v = 0; v < 8; ++v) {
        const int rr = v + 8 * half;
        G[rr        * G_STRIDE + r]       = g00[v];
        G[rr        * G_STRIDE + 16 + r]  = g01[v];
        G[(16 + r)  * G_STRIDE + rr]      = g01[v];   // transpose mirror = G10
        G[(16 + rr) * G_STRIDE + 16 + r]  = g11[v];
    }
    __syncthreads();

    // ---- Gauss-Jordan solve with register-resident rows (A SPD -> no pivoting). ----
    // Lane owns augmented row `lane` in 33 VGPRs; pivot row broadcast through LDS.
    float row[DF_ + 1];
    {
        const float4* myrow = reinterpret_cast<const float4*>(G + lane * G_STRIDE);
        #pragma unroll
        for (int q = 0; q < 8; ++q) {
            float4 p = myrow[q];
            row[4 * q + 0] = p.x;
            row[4 * q + 1] = p.y;
            row[4 * q + 2] = p.z;
            row[4 * q + 3] = p.w;
        }
        row[DF_] = G[lane * G_STRIDE + DF_];
    }

    float mydiag = 1.0f;
    #pragma unroll
    for (int k = 0; k < DF_; ++k) {
        if (lane == k) {   // publish current pivot row (9 LDS stores, single active lane)
            #pragma unroll
            for (int q = 0; q < 8; ++q) {
                reinterpret_cast<float4*>(pivrow)[q] =
                    make_float4(row[4 * q + 0], row[4 * q + 1], row[4 * q + 2], row[4 * q + 3]);
            }
            pivrow[DF_] = row[DF_];
        }
        __syncthreads();
        const float pk   = pivrow[k];         // broadcast read
        const float pinv = 1.0f / pk;
        float f = row[k] * pinv;
        if (lane == k) { mydiag = pk; f = 0.0f; }   // capture final diagonal; pivot row untouched
        #pragma unroll
        for (int q = 0; q < 8; ++q) {
            float4 p = reinterpret_cast<const float4*>(pivrow)[q];   // broadcast b128
            row[4 * q + 0] = fmaf(-f, p.x, row[4 * q + 0]);
            row[4 * q + 1] = fmaf(-f, p.y, row[4 * q + 1]);
            row[4 * q + 2] = fmaf(-f, p.z, row[4 * q + 2]);
            row[4 * q + 3] = fmaf(-f, p.w, row[4 * q + 3]);
        }
        row[DF_] = fmaf(-f, pivrow[DF_], row[DF_]);
        __syncthreads();
    }
    alpha[lane] = row[DF_] / mydiag;
    __syncthreads();

    // ---- o_hat[u] = sum_d alpha[d]*fp[d,u]; z[b,:] += beta[t]*o_hat ----
    const float bw = beta[t];
    float4 o = make_float4(0.0f, 0.0f, 0.0f, 0.0f);
    #pragma unroll 4
    for (int d = 0; d < DF_; ++d) {
        const float a = alpha[d];
        float4 f = *reinterpret_cast<const float4*>(fp_lds + d * FP_STRIDE + 4 * lane);
        o.x = fmaf(a, f.x, o.x);
        o.y = fmaf(a, f.y, o.y);
        o.z = fmaf(a, f.z, o.z);
        o.w = fmaf(a, f.w, o.w);
    }
    float* zb = z + (size_t)b * UNITS_ + 4 * lane;
    atomicAdd(zb + 0, bw * o.x);   // global_atomic_add_f32, L2-resident 128KB target
    atomicAdd(zb + 1, bw * o.y);
    atomicAdd(zb + 2, bw * o.z);
    atomicAdd(zb + 3, bw * o.w);
}

extern "C" void kernel_launch(void* const* d_in, const int* in_sizes, int n_in,
                              void* d_out, int out_size, void* d_ws, size_t ws_size,
                              hipStream_t stream) {
    (void)in_sizes; (void)n_in; (void)d_ws; (void)ws_size; (void)out_size;
    const float* inputs  = (const float*)d_in[0];  // (256,64,32)
    const float* latent  = (const float*)d_in[1];  // (256,64,128)
    const float* wkernel = (const float*)d_in[2];  // (1,1,32,128)
    const float* wbias   = (const float*)d_in[3];  // (1,1,32,128)
    const float* beta    = (const float*)d_in[4];  // (1,64,1)
    float* z = (float*)d_out;                      // (256,128)

    const int n = BS_ * UNITS_;
    hipLaunchKernelGGL(fta_zero_kernel, dim3((n + 255) / 256), dim3(256), 0, stream, z, n);
    hipLaunchKernelGGL(fta_kernel, dim3(BS_ * T_), dim3(32), 0, stream,
                       inputs, latent, wkernel, wbias, beta, z);
}